// MessageFactoryHelium2Bayonet_1228360646893
// MI455X (gfx1250) — compile-verified
//
#include <hip/hip_runtime.h>
#include <stdint.h>

#define N_SRC   100000
#define N_TGT   100000
#define N_EDGE  3200000
#define PI_F    3.14159265358979323846f
#define DIST_LO  0.0f
#define DIST_MID 0.001f

#define BLK             256
#define ITEMS           4
#define EDGES_PER_BLOCK (BLK * ITEMS)              // 1024
#define NBLK            (N_EDGE / EDGES_PER_BLOCK) // 3125, exact (no tail)

// d_out layout (flat, reference tuple order)
#define OUT_HEAT 0
#define OUT_MAXP 100000
#define OUT_BHX  100001
#define OUT_EVAP 200001
#define OUT_INMF 300001          // length N_EDGE+1
#define OUT_AVGF 3500002

__device__ __forceinline__ void atomAdd(float* p, float v) {
    // lowers to non-returning global_atomic_add_f32 (result unused)
    unsafeAtomicAdd(p, v);
}

__device__ __forceinline__ float edge_evap(float k, int ti,
                                           const float* __restrict__ L_tgt,
                                           const float* __restrict__ D_tgt,
                                           const float* __restrict__ F_tgt,
                                           float m1, float b1, float m2, float b2,
                                           float latent, float* f_out) {
    float D = D_tgt[ti];
    float L = L_tgt[ti];
    float f = F_tgt[ti];
    float circ   = PI_F * D;
    float creep  = fmaxf(fmaxf(m1 * k + b1, m2 * k + b2), 0.0f);
    float eff    = fminf((f * circ + 2.0f * creep) / circ, 1.0f);
    float wetted = L * D * PI_F * eff;
    *f_out = f;
    return (k * wetted) / latent;   // evap_mf
}

// ---------------- kernel 3 (placed first so disasm snippet shows the async path)
// main edge pass: async global->LDS stage + hierarchical scan + scatter
__global__ void k3_main(const float* __restrict__ kap,
                        const int*   __restrict__ src_idx,
                        const int*   __restrict__ tgt_idx,
                        const float* __restrict__ T_src,
                        const float* __restrict__ cp_src,
                        const float* __restrict__ T_tgt,
                        const float* __restrict__ L_tgt,
                        const float* __restrict__ D_tgt,
                        const float* __restrict__ F_tgt,
                        const float* __restrict__ latent_p,
                        const float* __restrict__ dt_p,
                        const float* __restrict__ lo_p,
                        const float* __restrict__ mid_p,
                        const float* __restrict__ hi_p,
                        const float* __restrict__ bo,
                        const float* __restrict__ total_ws,
                        float* __restrict__ out) {
    const int t = threadIdx.x, b = blockIdx.x;
    const int lane = t & 31, w = t >> 5;

    __shared__ float stage[EDGES_PER_BLOCK];   // 4KB: staged kapitza_hfd
    __shared__ float swave[BLK / 32];

    // --- CDNA5 async global->LDS stage of this block's kapitza stream (GVS mode).
    // Each wave fills its own contiguous 128-element LDS slice and only ever reads
    // its own slice afterwards, so a per-wave s_wait_asynccnt suffices (no barrier).
    const uint32_t lds0  = (uint32_t)(uintptr_t)(&stage[0]);  // low 32 bits = LDS offset
    const uint64_t gbase = (uint64_t)(uintptr_t)kap;
    #pragma unroll
    for (int i = 0; i < ITEMS; ++i) {
        int idxInBlk = w * 128 + i * 32 + lane;
        uint32_t laddr = lds0 + (uint32_t)idxInBlk * 4u;
        uint32_t goff  = (uint32_t)(b * EDGES_PER_BLOCK + idxInBlk) * 4u;
        asm volatile("global_load_async_to_lds_b32 %0, %1, %2"
                     :: "v"(laddr), "v"(goff), "s"(gbase) : "memory");
    }

    // scalar setup + index loads overlap with the in-flight async copies
    const float lo = lo_p[0], mid = mid_p[0], hi = hi_p[0];
    const float latent = latent_p[0], dt = dt_p[0];
    const float m1 = -(DIST_LO - DIST_MID) / (mid - lo);
    const float b1 = DIST_LO - m1 * lo;
    const float m2 = -DIST_MID / (hi - mid);
    const float b2 = -m2 * hi;
    const float total    = total_ws[0];
    const float blockoff = bo[b];

    const int vi = b * BLK + t;
    const int4 tv = reinterpret_cast<const int4*>(tgt_idx)[vi];
    const int4 sv = reinterpret_cast<const int4*>(src_idx)[vi];

    asm volatile("s_wait_asynccnt 0x0" ::: "memory");
    const float4 kv = reinterpret_cast<const float4*>(stage)[t]; // own wave's slice

    int   ti[ITEMS] = {tv.x, tv.y, tv.z, tv.w};
    int   si[ITEMS] = {sv.x, sv.y, sv.z, sv.w};
    float kk[ITEMS] = {kv.x, kv.y, kv.z, kv.w};
    float ev[ITEMS], av[ITEMS];
    #pragma unroll
    for (int i = 0; i < ITEMS; ++i)
        ev[i] = edge_evap(kk[i], ti[i], L_tgt, D_tgt, F_tgt, m1, b1, m2, b2, latent, &av[i]);

    // thread-local inclusive prefix
    float c0 = ev[0], c1 = c0 + ev[1], c2 = c1 + ev[2], c3 = c2 + ev[3];
    const float tsum = c3;

    // wave32 inclusive scan of thread sums (ds_bpermute shuffles)
    float x = tsum;
    #pragma unroll
    for (int off = 1; off < 32; off <<= 1) {
        float y = __shfl_up(x, off, 32);
        if (lane >= off) x += y;
    }
    if (lane == 31) swave[w] = x;          // wave total
    const float wexcl = x - tsum;
    __syncthreads();
    if (t == 0) {
        float run = 0.0f;
        #pragma unroll
        for (int j = 0; j < BLK / 32; ++j) { float sj = swave[j]; swave[j] = run; run += sj; }
    }
    __syncthreads();
    const float base_excl = blockoff + swave[w] + wexcl;  // global exclusive prefix

    const float excl[ITEMS] = {base_excl, base_excl + c0, base_excl + c1, base_excl + c2};
    const int e0 = vi * ITEMS;
    float* __restrict__ heat_pool = out + OUT_HEAT;
    float* __restrict__ evap_pool = out + OUT_EVAP;
    float* __restrict__ avgf_pool = out + OUT_AVGF;
    float* __restrict__ inmf_out  = out + OUT_INMF;

    #pragma unroll
    for (int i = 0; i < ITEMS; ++i) {
        float inc  = fmaxf(total - excl[i], 0.0f);        // incoming_mf[e]
        // 12.8MB write-once stream: non-temporal store keeps L2 for gather tables
        __builtin_nontemporal_store(inc, &inmf_out[e0 + i]);
        float curr = fminf(inc, ev[i]);                   // curr_evap_mf
        float energy = curr * latent * dt;
        float dT   = fmaxf(T_src[si[i]] - T_tgt[ti[i]], 0.0f);
        float maxE = dT * cp_src[si[i]];
        float heat = fminf(energy, maxE);
        atomAdd(&heat_pool[si[i]], heat);
        atomAdd(&evap_pool[ti[i]], curr);
        atomAdd(&avgf_pool[ti[i]], av[i]);
    }
}

// ---------------- kernel 0: zero the atomic pools ----------------
__global__ void k0_zero(float* __restrict__ out) {
    int i = blockIdx.x * blockDim.x + threadIdx.x;
    if (i < N_SRC)                out[OUT_HEAT + i] = 0.0f;
    else if (i < N_SRC + N_TGT)   out[OUT_EVAP + (i - N_SRC)] = 0.0f;
    else if (i < N_SRC + 2*N_TGT) out[OUT_AVGF + (i - N_SRC - N_TGT)] = 0.0f;
}

// ---------------- kernel 1: per-block partial sums of evap_mf ----------------
__global__ void k1_partials(const float* __restrict__ kap,
                            const int*   __restrict__ tgt_idx,
                            const float* __restrict__ L_tgt,
                            const float* __restrict__ D_tgt,
                            const float* __restrict__ F_tgt,
                            const float* __restrict__ latent_p,
                            const float* __restrict__ lo_p,
                            const float* __restrict__ mid_p,
                            const float* __restrict__ hi_p,
                            float* __restrict__ part) {
    const int t = threadIdx.x, b = blockIdx.x;
    const float lo = lo_p[0], mid = mid_p[0], hi = hi_p[0], latent = latent_p[0];
    const float m1 = -(DIST_LO - DIST_MID) / (mid - lo);
    const float b1 = DIST_LO - m1 * lo;
    const float m2 = -DIST_MID / (hi - mid);
    const float b2 = -m2 * hi;

    const int v = b * BLK + t;                       // vec4 index
    // RT (default) loads: kapitza is re-read by pass 2 and fits easily in 192MB L2
    const float4 kv = reinterpret_cast<const float4*>(kap)[v];
    const int4   tv = reinterpret_cast<const int4*>(tgt_idx)[v];

    float fdummy;
    float s = 0.0f;
    s += edge_evap(kv.x, tv.x, L_tgt, D_tgt, F_tgt, m1, b1, m2, b2, latent, &fdummy);
    s += edge_evap(kv.y, tv.y, L_tgt, D_tgt, F_tgt, m1, b1, m2, b2, latent, &fdummy);
    s += edge_evap(kv.z, tv.z, L_tgt, D_tgt, F_tgt, m1, b1, m2, b2, latent, &fdummy);
    s += edge_evap(kv.w, tv.w, L_tgt, D_tgt, F_tgt, m1, b1, m2, b2, latent, &fdummy);

    // wave32 reduce via lane shuffles
    #pragma unroll
    for (int off = 16; off > 0; off >>= 1) s += __shfl_down(s, off, 32);

    __shared__ float wsum[BLK / 32];
    if ((t & 31) == 0) wsum[t >> 5] = s;
    __syncthreads();
    if (t == 0) {
        float tot = 0.0f;
        #pragma unroll
        for (int j = 0; j < BLK / 32; ++j) tot += wsum[j];
        part[b] = tot;
    }
}

// ---------------- kernel 2: scan of block partials (single block) ----------------
#define K2T 256
#define K2C 13   // 256*13 = 3328 >= 3125

__global__ void k2_scan(const float* __restrict__ part,
                        float* __restrict__ bo,
                        const int*   __restrict__ tgt_idx,
                        const float* __restrict__ inmf_tgt,
                        const float* __restrict__ latent_p,
                        float* __restrict__ total_ws,
                        float* __restrict__ out) {
    const int t = threadIdx.x;
    __shared__ float ssum[K2T];
    __shared__ float s_total;
    if (t == 0) s_total = inmf_tgt[tgt_idx[0]];      // total_mf = prev_mf[0]

    float loc[K2C];
    float s = 0.0f;
    const int base = t * K2C;
    #pragma unroll
    for (int i = 0; i < K2C; ++i) {
        int idx = base + i;
        float v = (idx < NBLK) ? part[idx] : 0.0f;
        loc[i] = v;
        s += v;
    }
    ssum[t] = s;
    __syncthreads();

    // Hillis-Steele inclusive scan of 256 thread sums
    for (int off = 1; off < K2T; off <<= 1) {
        float v = (t >= off) ? ssum[t - off] : 0.0f;
        __syncthreads();
        ssum[t] += v;
        __syncthreads();
    }
    const float excl  = ssum[t] - s;
    const float grand = ssum[K2T - 1];
    const float total = s_total;

    if (t == 0) {
        out[OUT_MAXP] = total * latent_p[0];                 // max_power_sum
        total_ws[0]   = total;
        out[OUT_INMF + N_EDGE] = fmaxf(total - grand, 0.0f); // incoming_mf[-1]
    }
    float run = excl;
    #pragma unroll
    for (int i = 0; i < K2C; ++i) {
        int idx = base + i;
        if (idx < NBLK) bo[idx] = run;                       // exclusive block offset
        run += loc[i];
    }
}

// ---------------- kernel 4: power_to_bhx = heat_pool ----------------
__global__ void k4_copy(float* __restrict__ out) {
    int i = blockIdx.x * blockDim.x + threadIdx.x;
    if (i < N_SRC) out[OUT_BHX + i] = out[OUT_HEAT + i];
}

extern "C" void kernel_launch(void* const* d_in, const int* in_sizes, int n_in,
                              void* d_out, int out_size, void* d_ws, size_t ws_size,
                              hipStream_t stream) {
    const float* T_src    = (const float*)d_in[0];
    const float* cp_src   = (const float*)d_in[1];
    const float* T_tgt    = (const float*)d_in[2];
    const float* L_tgt    = (const float*)d_in[3];
    const float* D_tgt    = (const float*)d_in[4];
    const float* F_tgt    = (const float*)d_in[5];
    const float* inmf_tgt = (const float*)d_in[6];
    const float* kap      = (const float*)d_in[7];
    const int*   src_idx  = (const int*)d_in[8];
    const int*   tgt_idx  = (const int*)d_in[9];
    const float* latent   = (const float*)d_in[10];
    const float* dt       = (const float*)d_in[11];
    const float* lo       = (const float*)d_in[12];
    const float* mid      = (const float*)d_in[13];
    const float* hi       = (const float*)d_in[14];

    float* out      = (float*)d_out;
    float* ws       = (float*)d_ws;
    float* part     = ws;             // NBLK floats
    float* bo       = ws + NBLK;      // NBLK floats
    float* total_ws = ws + 2 * NBLK;  // 1 float

    k0_zero<<<(N_SRC + 2 * N_TGT + 255) / 256, 256, 0, stream>>>(out);
    k1_partials<<<NBLK, BLK, 0, stream>>>(kap, tgt_idx, L_tgt, D_tgt, F_tgt,
                                          latent, lo, mid, hi, part);
    k2_scan<<<1, K2T, 0, stream>>>(part, bo, tgt_idx, inmf_tgt, latent, total_ws, out);
    k3_main<<<NBLK, BLK, 0, stream>>>(kap, src_idx, tgt_idx, T_src, cp_src, T_tgt,
                                      L_tgt, D_tgt, F_tgt, latent, dt, lo, mid, hi,
                                      bo, total_ws, out);
    k4_copy<<<(N_SRC + 255) / 256, 256, 0, stream>>>(out);
}